// GCLSTM_14482629722194
// MI455X (gfx1250) — compile-verified
//
#include <hip/hip_runtime.h>
#include <hip/hip_bf16.h>

// ---------------------------------------------------------------------------
// GC-LSTM (Chebyshev graph conv LSTM) for MI455X / gfx1250, wave32 + WMMA.
//   N=50000 nodes, E=800000 edges, C=96 ch, T=4 steps, K=3 Cheb order.
// Dense math fused into one N x 384 x 384 fp32 WMMA GEMM per step:
//   gates = [Xt | H | T1 | T2] @ Wcat + bias,  Wcat built once from
//   Wp^T @ Wg (pool layer folded in) and cheb_W.
// A-panel staged into LDS with gfx1250 async global->LDS loads (ASYNCcnt).
// ---------------------------------------------------------------------------

#define NNODES 50000
#define NEDGES 800000
#define CH     96
#define TSTEPS 4
#define KCHEB  3
#define GCOLS  384              // 4 gates * 96
#define LDSA_STRIDE 388         // 384 + 4 pad: rows hit distinct LDS bank groups

typedef __attribute__((ext_vector_type(2))) float v2f;
typedef __attribute__((ext_vector_type(8))) float v8f;

// ---------------------------------------------------------------------------
// One-time weight preparation:
//   Wcat[r][col], r in [0,384): rows 0..95  -> (Wp^T @ Wg[g])[c][d]
//                               rows 96+k*96+c -> cheb_W[g][k][c][d]
//   col = g*96 + d.   bias[g*96+d] = bg[g][d] + cheb_b[g][d]
// ---------------------------------------------------------------------------
__global__ void prep_weights_kernel(const float* __restrict__ Wp,   // [C,C] (out,in)
                                    const float* __restrict__ Wg,   // [4,C,C] (in,out)
                                    const float* __restrict__ bg,   // [4,C]
                                    const float* __restrict__ chW,  // [4,K,C,C] (in,out)
                                    const float* __restrict__ chb,  // [4,C]
                                    float* __restrict__ Wcat,       // [384,384]
                                    float* __restrict__ bias)       // [384]
{
    int gid = blockIdx.x * blockDim.x + threadIdx.x;
    if (gid >= GCOLS * GCOLS) return;
    int r   = gid / GCOLS;       // input-channel row of A_concat
    int col = gid % GCOLS;
    int g   = col / CH;
    int d   = col % CH;
    float v;
    if (r < CH) {
        // Weff[g][c][d] = sum_e Wp[e][c] * Wg[g][e][d]
        int c = r;
        float acc = 0.f;
        for (int e = 0; e < CH; ++e)
            acc += Wp[e * CH + c] * Wg[(g * CH + e) * CH + d];
        v = acc;
    } else {
        int k = r / CH - 1;      // 0..2
        int c = r % CH;
        v = chW[((g * KCHEB + k) * CH + c) * CH + d];
    }
    Wcat[r * GCOLS + col] = v;
    if (gid < GCOLS) bias[gid] = bg[gid] + chb[gid];
}

// ---------------------------------------------------------------------------
// Small utility kernels
// ---------------------------------------------------------------------------
__global__ void zero_kernel(float* __restrict__ p, int n) {
    int i = blockIdx.x * blockDim.x + threadIdx.x;
    if (i < n) p[i] = 0.f;
}

__global__ void neg_copy_kernel(float* __restrict__ dst, const float* __restrict__ src, int n) {
    int i = blockIdx.x * blockDim.x + threadIdx.x;
    if (i < n) dst[i] = -src[i];            // T2 seed: -T0 (then scatter adds 2*prop(T1))
}

__global__ void degree_kernel(const int* __restrict__ srcI, const int* __restrict__ dstI,
                              float* __restrict__ deg, int ne) {
    int e = blockIdx.x * blockDim.x + threadIdx.x;
    if (e >= ne) return;
    int s = srcI[e], d = dstI[e];
    if (s != d) atomicAdd(&deg[d], 1.0f);   // edge weight 1, self-loops dropped
}

__global__ void dinv_kernel(float* __restrict__ deg, int n) {
    int i = blockIdx.x * blockDim.x + threadIdx.x;
    if (i >= n) return;
    float v = deg[i];
    deg[i] = (v > 0.f) ? rsqrtf(fmaxf(v, 1e-12f)) : 0.f;
}

// out[dst] += -scale * dinv[dst]*dinv[src] * x[src]   (lambda_max=2 => factor 1)
// 24 threads per edge, 4 channels each (float4 gather, 4 fp32 global atomics).
// Feature matrix (19.2 MB) is resident in the 192 MB L2, so gathers/atomics
// stay on-chip.
__global__ void prop_scatter_kernel(const float* __restrict__ x,
                                    float* __restrict__ outp,
                                    const float* __restrict__ dinv,
                                    const int* __restrict__ srcI,
                                    const int* __restrict__ dstI,
                                    float scale, int ne)
{
    int gid = blockIdx.x * blockDim.x + threadIdx.x;
    int e  = gid / (CH / 4);
    int cg = (gid % (CH / 4)) * 4;
    if (e >= ne) return;
    int s = srcI[e], d = dstI[e];
    if (s == d) return;
    float coef = -scale * dinv[d] * dinv[s];
    const float4 xv = *reinterpret_cast<const float4*>(x + (size_t)s * CH + cg);
    float* op = outp + (size_t)d * CH + cg;
    atomicAdd(op + 0, coef * xv.x);
    atomicAdd(op + 1, coef * xv.y);
    atomicAdd(op + 2, coef * xv.z);
    atomicAdd(op + 3, coef * xv.w);
}

// ---------------------------------------------------------------------------
// Fused gates GEMM (fp32 WMMA 16x16x4) + LSTM pointwise update.
// Block = 256 threads = 8 wave32, owns 16 node rows x all 384 gate columns.
//   - A panel [Xt|H|T1|T2] (16x384) staged to LDS via async global->LDS B128
//     copies (no VGPR round trip, drained with s_wait_asynccnt)
//   - wave w computes column tiles 3w..3w+2, K-loop of 96 v_wmma_f32_16x16x4_f32
//   - accumulators (+bias) staged back through LDS, then the block applies
//     sigmoid/tanh gates and writes H, Cstate, out[:,t,:]
// ---------------------------------------------------------------------------
__global__ __launch_bounds__(256) void gates_lstm_kernel(
        const float* __restrict__ Xt,    // [N,C]
        const float* __restrict__ Hst,   // [N,C]  (read as T0, rewritten at end)
        const float* __restrict__ Tx1,   // [N,C]
        const float* __restrict__ Tx2,   // [N,C]
        const float* __restrict__ Wcat,  // [384,384]
        const float* __restrict__ bias,  // [384]
        float* __restrict__ Hout,        // [N,C]  (== Hst buffer)
        float* __restrict__ Cst,         // [N,C]
        float* __restrict__ out,         // [N,T,C]
        int t)
{
    __shared__ float lds[16 * LDSA_STRIDE];
    const int tid = threadIdx.x;
    const int block_row = blockIdx.x * 16;

    // ---- stage A = [Xt | H | T1 | T2] rows (16 x 384 fp32) into LDS --------
    // Async copies: each lane moves one 16B chunk memory->LDS directly
    // (GLOBAL_LOAD_ASYNC_TO_LDS_B128, tracked by ASYNCcnt). LDS aperture puts
    // the 0-based LDS byte offset in the low 32 bits of the generic pointer.
    {
        const unsigned lds_base = (unsigned)(uintptr_t)(&lds[0]);
        for (int i = tid; i < 16 * (GCOLS / 4); i += 256) {   // 1536 float4 slots
            int row = i / (GCOLS / 4);
            int q4  = i % (GCOLS / 4);                        // float4 col index
            int blk = q4 / (CH / 4);                          // which source
            int c4  = q4 % (CH / 4);
            const float* sp = (blk == 0) ? Xt : (blk == 1) ? Hst : (blk == 2) ? Tx1 : Tx2;
            const float* gp = sp + (size_t)(block_row + row) * CH + c4 * 4;
            unsigned ldsaddr = lds_base +
                (unsigned)((row * LDSA_STRIDE + q4 * 4) * sizeof(float));   // 16B aligned
            asm volatile("global_load_async_to_lds_b128 %0, %1, off"
                         :: "v"(ldsaddr), "v"(gp) : "memory");
        }
        asm volatile("s_wait_asynccnt 0x0" ::: "memory");
    }
    __syncthreads();

    // ---- WMMA K-loop -------------------------------------------------------
    const int lane  = tid & 31;
    const int wave  = tid >> 5;
    const int m     = lane & 15;            // row (A) / col (B,C) within tile
    const int khalf = (lane >> 4) << 1;     // lanes 16-31 hold K+2,K+3

    v8f acc0 = {}, acc1 = {}, acc2 = {};
    const int cb0 = (wave * 3 + 0) * 16;
    const int cb1 = (wave * 3 + 1) * 16;
    const int cb2 = (wave * 3 + 2) * 16;

    for (int kk = 0; kk < GCOLS; kk += 4) {
        // A fragment: 16x4 fp32, lane holds A[m][kk+khalf .. +1] (8B-aligned LDS)
        const v2f a = *reinterpret_cast<const v2f*>(&lds[m * LDSA_STRIDE + kk + khalf]);
        const float* brow = Wcat + (size_t)(kk + khalf) * GCOLS + m;
        v2f b0; b0.x = brow[cb0]; b0.y = brow[cb0 + GCOLS];
        v2f b1; b1.x = brow[cb1]; b1.y = brow[cb1 + GCOLS];
        v2f b2; b2.x = brow[cb2]; b2.y = brow[cb2 + GCOLS];
        acc0 = __builtin_amdgcn_wmma_f32_16x16x4_f32(false, a, false, b0, (short)0, acc0, false, false);
        acc1 = __builtin_amdgcn_wmma_f32_16x16x4_f32(false, a, false, b1, (short)0, acc1, false, false);
        acc2 = __builtin_amdgcn_wmma_f32_16x16x4_f32(false, a, false, b2, (short)0, acc2, false, false);
    }
    __syncthreads();   // all A reads done; reuse LDS for gate pre-activations

    // ---- write accumulators (+bias) to LDS: lds[row][col] = gate preact ----
    {
        const int rbase = (lane >> 4) << 3;                   // C/D: M = r + 8*(lane>=16)
        float bv0 = bias[cb0 + m], bv1 = bias[cb1 + m], bv2 = bias[cb2 + m];
#pragma unroll
        for (int r = 0; r < 8; ++r) {
            int row = r + rbase;
            lds[row * LDSA_STRIDE + cb0 + m] = acc0[r] + bv0;
            lds[row * LDSA_STRIDE + cb1 + m] = acc1[r] + bv1;
            lds[row * LDSA_STRIDE + cb2 + m] = acc2[r] + bv2;
        }
    }
    __syncthreads();

    // ---- LSTM pointwise: gates order i, f, c, o ----------------------------
    for (int i = tid; i < 16 * CH; i += 256) {
        int row = i / CH, c = i % CH;
        size_t node = (size_t)(block_row + row);
        float gi = lds[row * LDSA_STRIDE + 0 * CH + c];
        float gf = lds[row * LDSA_STRIDE + 1 * CH + c];
        float gc = lds[row * LDSA_STRIDE + 2 * CH + c];
        float go = lds[row * LDSA_STRIDE + 3 * CH + c];
        float I  = 1.f / (1.f + __expf(-gi));
        float F  = 1.f / (1.f + __expf(-gf));
        float Tt = tanhf(gc);
        float O  = 1.f / (1.f + __expf(-go));
        float Cn = F * Cst[node * CH + c] + I * Tt;
        float Hn = O * tanhf(Cn);
        Cst[node * CH + c]  = Cn;
        Hout[node * CH + c] = Hn;
        out[node * (TSTEPS * CH) + t * CH + c] = Hn;
    }
}

// ---------------------------------------------------------------------------
// Host launch
// ---------------------------------------------------------------------------
extern "C" void kernel_launch(void* const* d_in, const int* in_sizes, int n_in,
                              void* d_out, int out_size, void* d_ws, size_t ws_size,
                              hipStream_t stream) {
    const float* X   = (const float*)d_in[0];   // [T,N,C]
    const int*   EI  = (const int*)  d_in[1];   // [T,2,E]
    const float* Wp  = (const float*)d_in[2];   // [C,C]
    const float* Wg  = (const float*)d_in[3];   // [4,C,C]
    const float* bg  = (const float*)d_in[4];   // [4,C]
    const float* chW = (const float*)d_in[5];   // [4,K,C,C]
    const float* chb = (const float*)d_in[6];   // [4,C]
    float* out = (float*)d_out;                 // [N,T,C]

    // workspace carve-up (floats): ~77.6 MB total
    float* ws   = (float*)d_ws;
    float* Wcat = ws;                              // 384*384
    float* bias = Wcat + GCOLS * GCOLS;            // 384
    float* dinv = bias + GCOLS;                    // N (deg -> dinv in place)
    float* Hst  = dinv + NNODES;                   // N*C
    float* Cstt = Hst  + (size_t)NNODES * CH;      // N*C
    float* Tx1  = Cstt + (size_t)NNODES * CH;      // N*C
    float* Tx2  = Tx1  + (size_t)NNODES * CH;      // N*C

    const int NC = NNODES * CH;
    dim3 blk(256);

    prep_weights_kernel<<<dim3((GCOLS * GCOLS + 255) / 256), blk, 0, stream>>>(
        Wp, Wg, bg, chW, chb, Wcat, bias);
    zero_kernel<<<dim3((NC + 255) / 256), blk, 0, stream>>>(Hst, NC);
    zero_kernel<<<dim3((NC + 255) / 256), blk, 0, stream>>>(Cstt, NC);

    for (int t = 0; t < TSTEPS; ++t) {
        const float* Xt   = X + (size_t)t * NNODES * CH;
        const int*   srcI = EI + (size_t)t * 2 * NEDGES;
        const int*   dstI = srcI + NEDGES;

        // symmetric normalization
        zero_kernel<<<dim3((NNODES + 255) / 256), blk, 0, stream>>>(dinv, NNODES);
        degree_kernel<<<dim3((NEDGES + 255) / 256), blk, 0, stream>>>(srcI, dstI, dinv, NEDGES);
        dinv_kernel<<<dim3((NNODES + 255) / 256), blk, 0, stream>>>(dinv, NNODES);

        // T1 = L_hat @ H
        zero_kernel<<<dim3((NC + 255) / 256), blk, 0, stream>>>(Tx1, NC);
        prop_scatter_kernel<<<dim3((NEDGES * (CH / 4) + 255) / 256), blk, 0, stream>>>(
            Hst, Tx1, dinv, srcI, dstI, 1.0f, NEDGES);

        // T2 = 2 * L_hat @ T1 - T0
        neg_copy_kernel<<<dim3((NC + 255) / 256), blk, 0, stream>>>(Tx2, Hst, NC);
        prop_scatter_kernel<<<dim3((NEDGES * (CH / 4) + 255) / 256), blk, 0, stream>>>(
            Tx1, Tx2, dinv, srcI, dstI, 2.0f, NEDGES);

        // fused gates GEMM (WMMA) + LSTM update + output write
        gates_lstm_kernel<<<dim3(NNODES / 16), blk, 0, stream>>>(
            Xt, Hst, Tx1, Tx2, Wcat, bias, Hst, Cstt, out, t);
    }
}